// BlockSparse_40845138985267
// MI455X (gfx1250) — compile-verified
//
#include <hip/hip_runtime.h>
#include <hip/hip_bf16.h>

typedef __attribute__((ext_vector_type(2))) float v2f;
typedef __attribute__((ext_vector_type(8))) float v8f;

#define BS      32
#define N_IN    4096
#define N_OUT   4096
#define N_BLK   4096
#define BATCH   1024
#define NCOLBLK (N_OUT / BS)   // 128 output block-columns
#define MAXPER  256            // >= max blocks per output column (mean 32, binomial)

// ---------------------------------------------------------------------------
// Kernel 0: build per-output-block-column lists (CSR by `co`) into workspace.
// 128 threads, one per output block-column. co is 16 KB -> L2-resident.
// ---------------------------------------------------------------------------
__global__ void bs_build_lists(const int* __restrict__ co,
                               int* __restrict__ lists,
                               int* __restrict__ counts) {
    const int t = threadIdx.x;  // 0..127
    int cnt = 0;
    for (int n = 0; n < N_BLK; ++n) {
        if (co[n] == t) {
            if (cnt < MAXPER) lists[t * MAXPER + cnt] = n;
            ++cnt;
        }
    }
    counts[t] = (cnt > MAXPER) ? MAXPER : cnt;
}

// ---------------------------------------------------------------------------
// Kernel 1: block-sparse GEMM via V_WMMA_F32_16X16X4_F32.
// grid = (128 out block-cols, 8 row tiles), block = 256 threads = 8 waves.
// Each wave computes a 16x32 output tile (2 accumulators of v8f).
//
// Weight block staged double-buffered in LDS in *fragment-packed* order:
//   element (k, n) of the 32x32 block lives at  (k>>1)*64 + 2*n + (k&1)
// so a B fragment (two consecutive K for one N) is a single ds_load_b64.
// Pipeline: store regs -> issue next block's global load -> barrier -> WMMA.
// ---------------------------------------------------------------------------
__global__ __launch_bounds__(256) void bs_wmma_gemm(
        const float* __restrict__ x,      // (BATCH, N_IN)
        const float* __restrict__ kern,   // (N_BLK, 32, 32)
        const float* __restrict__ bias,   // (N_OUT)
        const int*   __restrict__ ci,     // (N_BLK)
        const int*   __restrict__ lists,  // (NCOLBLK, MAXPER)
        const int*   __restrict__ counts, // (NCOLBLK)
        float* __restrict__ out) {        // (BATCH, N_OUT)
    __shared__ float Bs[2][BS * BS];      // double buffer, 2 x 4 KB

    const int obk  = blockIdx.x;          // output block column 0..127
    const int tid  = threadIdx.x;
    const int wave = tid >> 5;            // 0..7
    const int lane = tid & 31;
    const int half = lane >> 4;           // 0/1: lane half-group
    const int l    = lane & 15;

    const int rowBase = blockIdx.y * 128 + wave * 16;  // this wave's 16 rows

    // Staging coords: 2x2 patch (rows 2*i2, 2*i2+1; cols 2*o2, 2*o2+1).
    const int i2 = tid >> 4;              // 0..15
    const int o2 = tid & 15;              // 0..15

    v8f acc0 = {};  // output cols [obk*32 + 0 .. 15]
    v8f acc1 = {};  // output cols [obk*32 + 16 .. 31]

    const int cnt = counts[obk];

    // Prologue: load block 0's weight patch into registers.
    v2f r0 = {}, r1 = {};
    if (cnt > 0) {
        const int n0 = lists[obk * MAXPER];
        const float* src = kern + (size_t)n0 * (BS * BS) + (2 * i2) * BS + 2 * o2;
        r0 = *(const v2f*)(src);
        r1 = *(const v2f*)(src + BS);
    }

    for (int i = 0; i < cnt; ++i) {
        const int par = i & 1;
        const int n   = lists[obk * MAXPER + i];
        const int cib = ci[n];

        // Store pre-loaded patch into LDS, fragment-packed:
        // (k,n) -> (k>>1)*64 + 2*n + (k&1); 2x2 patch lands at 4*tid.
        float4 packed;
        packed.x = r0.x; packed.y = r1.x; packed.z = r0.y; packed.w = r1.y;
        *(float4*)(&Bs[par][4 * tid]) = packed;

        // Issue next block's global load: latency hides under this block's WMMAs.
        if (i + 1 < cnt) {
            const int n2 = lists[obk * MAXPER + i + 1];
            const float* src = kern + (size_t)n2 * (BS * BS) + (2 * i2) * BS + 2 * o2;
            r0 = *(const v2f*)(src);
            r1 = *(const v2f*)(src + BS);
            if (i + 2 < cnt) {
                const int n3 = lists[obk * MAXPER + i + 2];
                __builtin_prefetch(kern + (size_t)n3 * (BS * BS) + tid * 4, 0, 0);
            }
        }

        __syncthreads();  // Bs[par] ready; prior readers of Bs[par] (iter i-2)
                          // completed before the barrier in iter i-1.

        // A rows: x[rowBase + l][cib*32 + k]. ISA 16x4 f32 A layout:
        // VGPR v holds K = 2*half + v, M = l -> one b64 load per k-step.
        const float* xrow = x + (size_t)(rowBase + l) * N_IN + cib * BS;
        const float* bsl  = &Bs[par][0];

        #pragma unroll
        for (int kk = 0; kk < BS; kk += 4) {
            const v2f a = *(const v2f*)(xrow + kk + 2 * half);
            // Fragment-packed B: one contiguous b64 per fragment.
            const int kb = ((kk >> 1) + half) * 64;
            const v2f b0 = *(const v2f*)(bsl + kb + 2 * l);
            const v2f b1 = *(const v2f*)(bsl + kb + 2 * (16 + l));
            acc0 = __builtin_amdgcn_wmma_f32_16x16x4_f32(
                false, a, false, b0, (short)0, acc0, false, false);
            acc1 = __builtin_amdgcn_wmma_f32_16x16x4_f32(
                false, a, false, b1, (short)0, acc1, false, false);
        }
    }

    // Epilogue: bias + ReLU. C/D layout: VGPR v -> row v + 8*half, col = l (+16).
    const int colBase = obk * BS;
    const float bia0 = bias[colBase + l];
    const float bia1 = bias[colBase + 16 + l];
    #pragma unroll
    for (int v = 0; v < 8; ++v) {
        const int r = rowBase + v + 8 * half;
        float o0 = acc0[v] + bia0;
        float o1 = acc1[v] + bia1;
        o0 = o0 > 0.0f ? o0 : 0.0f;
        o1 = o1 > 0.0f ? o1 : 0.0f;
        out[(size_t)r * N_OUT + colBase + l]      = o0;
        out[(size_t)r * N_OUT + colBase + 16 + l] = o1;
    }
}

// ---------------------------------------------------------------------------
// Launch: inputs in setup_inputs() order: x, kernel, bias, ci, co.
// ---------------------------------------------------------------------------
extern "C" void kernel_launch(void* const* d_in, const int* in_sizes, int n_in,
                              void* d_out, int out_size, void* d_ws, size_t ws_size,
                              hipStream_t stream) {
    const float* x    = (const float*)d_in[0];
    const float* kern = (const float*)d_in[1];
    const float* bias = (const float*)d_in[2];
    const int*   ci   = (const int*)d_in[3];
    const int*   co   = (const int*)d_in[4];
    float*       out  = (float*)d_out;

    int* lists  = (int*)d_ws;                       // 128 * 256 ints = 128 KB
    int* counts = lists + NCOLBLK * MAXPER;         // 128 ints

    bs_build_lists<<<1, NCOLBLK, 0, stream>>>(co, lists, counts);

    dim3 grid(NCOLBLK, BATCH / 128);
    bs_wmma_gemm<<<grid, 256, 0, stream>>>(x, kern, bias, ci, lists, counts, out);
}